// PILCO_22462678958631
// MI455X (gfx1250) — compile-verified
//
#include <hip/hip_runtime.h>
#include <hip/hip_bf16.h>

// ---------------------------------------------------------------------------
// PILCO moment-matched rollout for MI455X (gfx1250, wave32).
// Latency-bound problem: all state lives in LDS of one WGP for the rollout;
// f32 WMMA 16x16x4 used for the per-pair N x D matmuls.
// ---------------------------------------------------------------------------

#define NTRAIN 100
#define NPAD   112        // 7 tiles of 16
#define DDIM   16
#define EDIM   12
#define UDIM   4
#define NPAIR  78         // E*(E+1)/2
#define NW     8          // waves per block (256 threads)
#define JITTER 1e-4f

typedef float v2f __attribute__((ext_vector_type(2)));
typedef float v8f __attribute__((ext_vector_type(8)));

__device__ __forceinline__ v8f wmma4(v2f a, v2f b, v8f c) {
  // D = A(16x4,f32) * B(4x16,f32) + C(16x16,f32)
  return __builtin_amdgcn_wmma_f32_16x16x4_f32(false, a, false, b,
                                               (short)0, c, false, false);
}

// In-order LDS per wave + lockstep wave32 makes cross-lane LDS communication
// safe provided the compiler neither caches nor reorders across this point.
__device__ __forceinline__ void lds_fence() {
  asm volatile("s_wait_dscnt 0" ::: "memory");
}

__device__ __forceinline__ float wave_sum(float v) {
  #pragma unroll
  for (int m = 16; m >= 1; m >>= 1) v += __shfl_xor(v, m, 32);
  return v;
}

// Gauss-Jordan inverse of an n x n system stored as [A | I] (n rows, 2n cols)
// in LDS; one wave, lane owns column `lane` (2n <= 32). Returns det(A) on all
// lanes. SPD inputs -> no pivoting needed.
__device__ float wave_gj(float* A, int n, int lane) {
  const int w = 2 * n;
  float det = 1.0f;
  #pragma unroll 1
  for (int p = 0; p < n; p++) {
    lds_fence();
    float piv = A[p * w + p];
    det *= piv;
    float ip = 1.0f / piv;
    if (lane < w) A[p * w + lane] *= ip;
    lds_fence();
    float prow = (lane < w) ? A[p * w + lane] : 0.0f;
    #pragma unroll 1
    for (int r = 0; r < n; r++) {
      if (r == p) continue;
      float f = A[r * w + p];          // pre-update value of row r, col p
      if (lane < w) A[r * w + lane] -= f * prow;
    }
  }
  lds_fence();
  return det;
}

// ---------------------------------------------------------------------------
// Kernel 1: build augmented [Ky | I] systems, one block per output e.
// ws layout: aug[e][r][c], r<100, c<200.
// ---------------------------------------------------------------------------
__global__ void pilco_prep(const float* __restrict__ X,
                           const float* __restrict__ ls,
                           const float* __restrict__ var_,
                           const float* __restrict__ noise,
                           float* __restrict__ aug) {
  __shared__ float sX[NTRAIN * DDIM];
  __shared__ float siL[DDIM];
  const int e = blockIdx.x;
  const int t = threadIdx.x;
  for (int i = t; i < NTRAIN * DDIM; i += 256) sX[i] = X[i];
  if (t < DDIM) siL[t] = 1.0f / ls[e * DDIM + t];
  __syncthreads();
  const float vn = var_[e];
  const float nj = noise[e] + JITTER;
  #pragma unroll 1
  for (int idx = t; idx < NTRAIN * NTRAIN; idx += 256) {
    int r = idx / NTRAIN, c = idx % NTRAIN;
    float d2 = 0.0f;
    #pragma unroll
    for (int d = 0; d < DDIM; d++) {
      float df = (sX[r * DDIM + d] - sX[c * DDIM + d]) * siL[d];
      d2 += df * df;
    }
    float k = vn * __expf(-0.5f * d2) + ((r == c) ? nj : 0.0f);
    float* A = aug + ((size_t)e * NTRAIN + r) * 200;
    A[c] = k;
    A[100 + c] = (r == c) ? 1.0f : 0.0f;
  }
}

// ---------------------------------------------------------------------------
// Kernel 2: in-place Gauss-Jordan; right half becomes iK. One block per e.
// ---------------------------------------------------------------------------
__global__ void pilco_invert(float* __restrict__ aug) {
  __shared__ float prow[200];
  __shared__ float ipv;
  const int e = blockIdx.x, t = threadIdx.x;
  float* A = aug + (size_t)e * NTRAIN * 200;
  #pragma unroll 1
  for (int p = 0; p < NTRAIN; p++) {
    if (t == 0) ipv = 1.0f / A[p * 200 + p];
    __syncthreads();
    for (int c = t; c < 200; c += 256) {
      float v = A[p * 200 + c] * ipv;
      prow[c] = v;
      A[p * 200 + c] = v;
    }
    __syncthreads();
    #pragma unroll 1
    for (int r = t; r < NTRAIN; r += 256) {
      if (r == p) continue;
      float f = A[r * 200 + p];
      #pragma unroll 1
      for (int c = 0; c < 200; c++) A[r * 200 + c] -= f * prow[c];
    }
    __syncthreads();
  }
}

// ---------------------------------------------------------------------------
// Kernel 3: persistent single-block rollout (256 threads = 8 wave32s).
// ---------------------------------------------------------------------------
__global__ void __launch_bounds__(256)
pilco_rollout(const float* __restrict__ Xg, const float* __restrict__ Yg,
              const float* __restrict__ lsg, const float* __restrict__ varg,
              const float* __restrict__ wcg, const float* __restrict__ bcg,
              const float* __restrict__ m0g, const float* __restrict__ S0g,
              const int* __restrict__ horg, const float* __restrict__ iKg,
              float* __restrict__ out) {
  __shared__ float inp[NPAD][DDIM];          // X - m_cat (padded rows = 0)
  __shared__ float iLs[EDIM][DDIM];
  __shared__ float il2[EDIM][DDIM];
  __shared__ float var_s[EDIM], logvar[EDIM];
  __shared__ float wc[UDIM][EDIM], bc[UDIM];
  __shared__ float m_state[EDIM], s_state[EDIM][EDIM];
  __shared__ float m_cat[DDIM], s_cat[DDIM][DDIM];
  __shared__ float sV[EDIM][UDIM];
  __shared__ float betap[EDIM][NPAD];
  __shared__ float kk[EDIM][NPAD];
  __shared__ float Mdx[EDIM], Vdx[EDIM][DDIM], Sdx[EDIM][EDIM];
  __shared__ float Mx[EDIM], Sx[EDIM][EDIM], s1C[EDIM][EDIM];
  __shared__ float Aaug[EDIM * 24];
  __shared__ float r_acc;
  __shared__ float wscr[NW][768];            // per-wave scratch (3 KB each)

  const int t = threadIdx.x;
  const int lane = t & 31;
  const int wid = t >> 5;
  const int half = lane >> 4;
  const int l15 = lane & 15;

  // ---- one-time init -------------------------------------------------------
  for (int i = t; i < EDIM * DDIM; i += 256) {
    float il = 1.0f / lsg[i];
    iLs[i / DDIM][i % DDIM] = il;
    il2[i / DDIM][i % DDIM] = il * il;
  }
  if (t < EDIM) {
    var_s[t] = varg[t];
    logvar[t] = __logf(varg[t]);
    m_state[t] = m0g[t];
  }
  if (t < UDIM * EDIM) ((float*)wc)[t] = wcg[t];
  if (t < UDIM) bc[t] = bcg[t];
  for (int i = t; i < EDIM * EDIM; i += 256) s_state[i / EDIM][i % EDIM] = S0g[i];
  #pragma unroll 1
  for (int i = t; i < EDIM * NPAD; i += 256) {
    int e = i / NPAD, n = i % NPAD;
    float b = 0.0f;
    if (n < NTRAIN) {
      const float* row = iKg + ((size_t)e * NTRAIN + n) * 200 + 100;
      #pragma unroll 1
      for (int m = 0; m < NTRAIN; m++) b += row[m] * Yg[m * EDIM + e];
    }
    betap[e][n] = b;
  }
  if (t == 0) r_acc = 0.0f;
  const int H = horg[0];
  __syncthreads();

  // ---- horizon loop --------------------------------------------------------
  #pragma unroll 1
  for (int step = 0; step < H; step++) {
    // Phase 0a: controller pieces
    if (t < EDIM * UDIM) {                   // sV = s @ Wc^T  (12x4)
      int a = t / UDIM, u = t % UDIM;
      float acc = 0.0f;
      for (int k = 0; k < EDIM; k++) acc += s_state[a][k] * wc[u][k];
      sV[a][u] = acc;
    }
    if (t >= 64 && t < 64 + UDIM) {          // m_u
      int u = t - 64;
      float acc = bc[u];
      for (int k = 0; k < EDIM; k++) acc += m_state[k] * wc[u][k];
      m_cat[EDIM + u] = acc;
    }
    if (t >= 96 && t < 96 + EDIM) m_cat[t - 96] = m_state[t - 96];
    __syncthreads();

    // Phase 0b: s_cat assembly + inp = X - m_cat (zero-padded)
    {
      int a = t >> 4, b = t & 15;            // 256 threads = all entries
      float v;
      if (a < EDIM && b < EDIM)      v = s_state[a][b];
      else if (a < EDIM)             v = sV[a][b - EDIM];
      else if (b < EDIM)             v = sV[b][a - EDIM];
      else {
        float acc = 0.0f;
        for (int k = 0; k < EDIM; k++) acc += wc[a - EDIM][k] * sV[k][b - EDIM];
        v = acc;                             // s_u = Wc s Wc^T
      }
      s_cat[a][b] = v;
    }
    #pragma unroll 1
    for (int idx = t; idx < NPAD * DDIM; idx += 256) {
      int n = idx / DDIM, d = idx % DDIM;
      inp[n][d] = (n < NTRAIN) ? (Xg[n * DDIM + d] - m_cat[d]) : 0.0f;
    }
    __syncthreads();

    // Phase 1+2: per-output B inverse, lb/kk, M and V   (wave per e)
    #pragma unroll 1
    for (int e = wid; e < EDIM; e += NW) {
      float* augp = &wscr[wid][0];           // [16][32] = [B | I]
      #pragma unroll 1
      for (int r = 0; r < DDIM; r++) {
        float v;
        if (lane < DDIM)
          v = iLs[e][r] * s_cat[r][lane] * iLs[e][lane] + ((r == lane) ? 1.0f : 0.0f);
        else
          v = ((lane - DDIM) == r) ? 1.0f : 0.0f;
        augp[r * 32 + lane] = v;
      }
      float detB = wave_gj(augp, DDIM, lane);
      float cE = var_s[e] * rsqrtf(detB);
      float macc = 0.0f, vacc[DDIM];
      #pragma unroll
      for (int d = 0; d < DDIM; d++) vacc[d] = 0.0f;
      #pragma unroll 1
      for (int n = lane; n < NPAD; n += 32) {
        float iNv[DDIM], sum2 = 0.0f;
        #pragma unroll
        for (int d = 0; d < DDIM; d++) {
          iNv[d] = inp[n][d] * iLs[e][d];
          sum2 += iNv[d] * iNv[d];
        }
        float snt = 0.0f;
        #pragma unroll 1
        for (int d = 0; d < DDIM; d++) {     // pass 1: snt = iN . (iN @ iB)
          float td = 0.0f;
          for (int d2 = 0; d2 < DDIM; d2++) td += iNv[d2] * augp[d2 * 32 + 16 + d];
          snt += iNv[d] * td;
        }
        kk[e][n] = logvar[e] - 0.5f * sum2;
        float w = __expf(-0.5f * snt) * betap[e][n];
        macc += w;
        #pragma unroll 1
        for (int d = 0; d < DDIM; d++) {     // pass 2: recompute t_d for V
          float td = 0.0f;
          for (int d2 = 0; d2 < DDIM; d2++) td += iNv[d2] * augp[d2 * 32 + 16 + d];
          vacc[d] += td * iLs[e][d] * w;
        }
      }
      macc = wave_sum(macc);
      #pragma unroll 1
      for (int d = 0; d < DDIM; d++) vacc[d] = wave_sum(vacc[d]);
      if (lane == 0) {
        Mdx[e] = cE * macc;
        for (int d = 0; d < DDIM; d++) Vdx[e][d] = cE * vacc[d];
      }
    }
    __syncthreads();

    // Phase 3: pair loop (i<=j), wave per pair, WMMA tiles
    #pragma unroll 1
    for (int p = wid; p < NPAIR; p += NW) {
      int i = 0, pp = p;
      while (pp >= EDIM - i) { pp -= (EDIM - i); i++; }
      int j = i + pp;
      float* augp = &wscr[wid][0];           // [16][32] = [R | I]
      float* Qp = &wscr[wid][512];           // [16][16]
      #pragma unroll 1
      for (int r = 0; r < DDIM; r++) {
        float v;
        if (lane < DDIM)
          v = s_cat[r][lane] * (il2[i][lane] + il2[j][lane]) + ((r == lane) ? 1.0f : 0.0f);
        else
          v = ((lane - DDIM) == r) ? 1.0f : 0.0f;
        augp[r * 32 + lane] = v;
      }
      float detR = wave_gj(augp, DDIM, lane);
      lds_fence();
      // Q = 0.5 * iR @ s_cat  (WMMA, K in 4 chunks of 4)
      v8f cq = {};
      #pragma unroll
      for (int kc = 0; kc < 4; kc++) {
        int kb = kc * 4 + 2 * half;
        v2f a, b;
        a.x = augp[l15 * 32 + 16 + kb];
        a.y = augp[l15 * 32 + 16 + kb + 1];
        b.x = s_cat[kb][l15];
        b.y = s_cat[kb + 1][l15];
        cq = wmma4(a, b, cq);
      }
      #pragma unroll
      for (int v = 0; v < 8; v++) Qp[(v + 8 * half) * 16 + l15] = 0.5f * cq[v];
      lds_fence();
      // ai[n] = kk_i[n] + Xa_i Q Xa_i ; bj[m] = kk_j[m] + Xa_j Q Xa_j
      float* Pt  = &wscr[wid][0];            // reuse aug region
      float* aip = &wscr[wid][256];
      float* bjp = &wscr[wid][368];
      #pragma unroll 1
      for (int n = lane; n < NPAD; n += 32) {
        float xa[DDIM], xb[DDIM];
        #pragma unroll
        for (int d = 0; d < DDIM; d++) {
          xa[d] = inp[n][d] * il2[i][d];
          xb[d] = inp[n][d] * il2[j][d];
        }
        float xsa = 0.0f, xsb = 0.0f;
        #pragma unroll 1
        for (int d2 = 0; d2 < DDIM; d2++) {
          float ta = 0.0f, tb = 0.0f;
          for (int d = 0; d < DDIM; d++) {
            float q = Qp[d * 16 + d2];
            ta += xa[d] * q;
            tb += xb[d] * q;
          }
          xsa += ta * xa[d2];
          xsb += tb * xb[d2];
        }
        aip[n] = kk[i][n] + xsa;
        bjp[n] = kk[j][n] + xsb;
      }
      lds_fence();
      float accS = 0.0f, accT = 0.0f;
      #pragma unroll 1
      for (int ti = 0; ti < 7; ti++) {
        v8f cP = {};                          // P = Xa_i @ Q (one 16-row tile)
        #pragma unroll
        for (int kc = 0; kc < 4; kc++) {
          int kb = kc * 4 + 2 * half;
          int row = ti * 16 + l15;
          v2f a, b;
          a.x = inp[row][kb]     * il2[i][kb];
          a.y = inp[row][kb + 1] * il2[i][kb + 1];
          b.x = Qp[kb * 16 + l15];
          b.y = Qp[(kb + 1) * 16 + l15];
          cP = wmma4(a, b, cP);
        }
        #pragma unroll
        for (int v = 0; v < 8; v++) Pt[(v + 8 * half) * 16 + l15] = cP[v];
        lds_fence();
        int rowbase = ti * 16 + 8 * half;
        #pragma unroll 1
        for (int tj = 0; tj < 7; tj++) {
          v8f cC = {};                        // P @ Xa_j^T (16x16 tile)
          #pragma unroll
          for (int kc = 0; kc < 4; kc++) {
            int kb = kc * 4 + 2 * half;
            int mcol = tj * 16 + l15;
            v2f a, b;
            a.x = Pt[l15 * 16 + kb];
            a.y = Pt[l15 * 16 + kb + 1];
            b.x = inp[mcol][kb]     * il2[j][kb];
            b.y = inp[mcol][kb + 1] * il2[j][kb + 1];
            cC = wmma4(a, b, cC);
          }
          int col = tj * 16 + l15;
          float bjc  = bjp[col];
          float betj = betap[j][col];
          #pragma unroll
          for (int v = 0; v < 8; v++) {
            int row = rowbase + v;
            float eV = __expf(2.0f * cC[v] + aip[row] + bjc);
            accS += betap[i][row] * betj * eV;
            if (i == j && row < NTRAIN && col < NTRAIN)
              accT += iKg[((size_t)i * NTRAIN + row) * 200 + 100 + col] * eV;
          }
        }
        lds_fence();
      }
      accS = wave_sum(accS);
      accT = wave_sum(accT);
      if (lane == 0) {
        float sc = rsqrtf(detR) * var_s[i] * var_s[j];
        float val = (accS - accT) * sc;
        Sdx[i][j] = val;
        Sdx[j][i] = val;
      }
    }
    __syncthreads();

    // Phase 4a: s1C = s_cat[0:12,:] @ V^T ; Mx = Mdx + m
    if (t < EDIM * EDIM) {
      int a = t / EDIM, b = t % EDIM;
      float acc = 0.0f;
      for (int d = 0; d < DDIM; d++) acc += s_cat[a][d] * Vdx[b][d];
      s1C[a][b] = acc;
    }
    if (t >= 192 && t < 192 + EDIM) Mx[t - 192] = Mdx[t - 192] + m_state[t - 192];
    __syncthreads();

    // Phase 4b: Sx = Sdx - M M^T + s + s1C + s1C^T
    if (t < EDIM * EDIM) {
      int a = t / EDIM, b = t % EDIM;
      Sx[a][b] = Sdx[a][b] - Mdx[a] * Mdx[b] + s_state[a][b] + s1C[a][b] + s1C[b][a];
    }
    __syncthreads();

    // Phase 5: reward (wave 0) + state update (other threads)
    if (wid == 0) {
      #pragma unroll 1
      for (int r = 0; r < EDIM; r++) {
        if (lane < 24) {
          float v = (lane < EDIM) ? (Sx[r][lane] + ((lane == r) ? 1.0f : 0.0f))
                                  : (((lane - EDIM) == r) ? 1.0f : 0.0f);
          Aaug[r * 24 + lane] = v;
        }
      }
      float detA = wave_gj(Aaug, EDIM, lane);
      lds_fence();
      float part = 0.0f;
      if (lane < EDIM) {
        float s = 0.0f;
        for (int a = 0; a < EDIM; a++) s += Mx[a] * Aaug[a * 24 + EDIM + lane];
        part = s * Mx[lane];
      }
      float q = wave_sum(part);
      if (lane == 0) r_acc += __expf(-0.5f * q) * rsqrtf(detA);
    }
    if (t >= 32 && t < 32 + EDIM) m_state[t - 32] = Mx[t - 32];
    if (t >= 64 && t < 64 + EDIM * EDIM) {
      int idx = t - 64;
      s_state[idx / EDIM][idx % EDIM] = Sx[idx / EDIM][idx % EDIM];
    }
    __syncthreads();
  }

  // ---- output: [m(12), s(144), r(1)] --------------------------------------
  if (t < EDIM) out[t] = m_state[t];
  if (t >= 32 && t < 32 + EDIM * EDIM) {
    int idx = t - 32;
    out[EDIM + idx] = s_state[idx / EDIM][idx % EDIM];
  }
  if (t == 0) out[EDIM + EDIM * EDIM] = r_acc;
}

// ---------------------------------------------------------------------------
extern "C" void kernel_launch(void* const* d_in, const int* in_sizes, int n_in,
                              void* d_out, int out_size, void* d_ws, size_t ws_size,
                              hipStream_t stream) {
  const float* X     = (const float*)d_in[0];
  const float* Y     = (const float*)d_in[1];
  const float* ls    = (const float*)d_in[2];
  const float* var_  = (const float*)d_in[3];
  const float* noise = (const float*)d_in[4];
  const float* Wc    = (const float*)d_in[5];
  const float* bc    = (const float*)d_in[6];
  const float* m0    = (const float*)d_in[7];
  const float* S0    = (const float*)d_in[8];
  const int*   hor   = (const int*)d_in[9];
  float* out = (float*)d_out;
  float* aug = (float*)d_ws;   // 12 * 100 * 200 floats = 960 KB of scratch

  pilco_prep  <<<dim3(12), 256, 0, stream>>>(X, ls, var_, noise, aug);
  pilco_invert<<<dim3(12), 256, 0, stream>>>(aug);
  pilco_rollout<<<dim3(1), 256, 0, stream>>>(X, Y, ls, var_, Wc, bc, m0, S0,
                                             hor, aug, out);
}